// SparseAttention_4990751998151
// MI455X (gfx1250) — compile-verified
//
#include <hip/hip_runtime.h>
#include <cstdint>

// ---------------------------------------------------------------------------
// SparseAttention (no softmax) on MI455X / gfx1250.
//
//   out = (Q K^T * scale) @ V  ==  scale * Q @ (K^T @ V)     (associativity)
//
// Phase 1: 512 blocks (64 heads x 8 t-segments) compute partial M = K^T V
//          per segment with V_WMMA_F32_16X16X4_F32, staged through LDS with
//          double-buffered GLOBAL_LOAD_ASYNC_TO_LDS_B128 (ASYNCcnt-tracked).
// Phase 2: 2048 blocks reduce the 8 L2-resident partials (fixed order ->
//          deterministic), scale, and compute out = Q @ M with WMMA.
// Traffic floor: ~134 MB @ 23.3 TB/s ~ 6 us; math is exact f32 WMMA.
// ---------------------------------------------------------------------------

#define SEQ   2048
#define DIM   64
#define NHEAD 64          // B*H = 4*16
#define LSTR  65          // padded stride for scalar-staged buffers
#define KSTR  72          // 16B-aligned rows for async b128; rows r/r+2 differ
                          // by 16 banks -> half-wave frag reads conflict-free

typedef float v2f __attribute__((ext_vector_type(2)));
typedef float v8f __attribute__((ext_vector_type(8)));
typedef int   v4i __attribute__((ext_vector_type(4)));

typedef __attribute__((address_space(1))) v4i gv4i;   // global b128
typedef __attribute__((address_space(3))) v4i lv4i;   // LDS b128

#if __has_builtin(__builtin_amdgcn_global_load_async_to_lds_b128) && \
    __has_builtin(__builtin_amdgcn_s_wait_asynccnt)
#define USE_ASYNC_LDS 1
#else
#define USE_ASYNC_LDS 0
#endif

__device__ __forceinline__ v8f wmma4_f32(v2f a, v2f b, v8f c) {
  // (neg_a, A, neg_b, B, c_mod, C, reuse_a, reuse_b)
  return __builtin_amdgcn_wmma_f32_16x16x4_f32(false, a, false, b, (short)0, c,
                                               false, false);
}

// Issue staging of one 64x64 f32 tile into LDS (row stride KSTR).
// Async path: 4 x b128 async-to-LDS per thread (8 wave instructions per
// K+V chunk). Fallback: float4 global load + scalar LDS stores.
__device__ __forceinline__ void stage_issue(const float* __restrict__ g,
                                            float* __restrict__ lds, int tid) {
#if USE_ASYNC_LDS
#pragma unroll
  for (int i = 0; i < 4; ++i) {
    const int e   = tid + i * 256;   // float4 slot 0..1023
    const int row = e >> 4;
    const int c4  = (e & 15) * 4;
    gv4i* gp = (gv4i*)(uintptr_t)(g + e * 4);
    lv4i* lp = (lv4i*)(uintptr_t)(lds + row * KSTR + c4);
    __builtin_amdgcn_global_load_async_to_lds_b128(gp, lp, 0, 0);
  }
#else
  const float4* g4 = (const float4*)g;
#pragma unroll
  for (int i = 0; i < 4; ++i) {
    const int e   = tid + i * 256;
    const int row = e >> 4;
    const int c4  = (e & 15) * 4;
    float4 x = g4[e];
    float* d = &lds[row * KSTR + c4];
    d[0] = x.x; d[1] = x.y; d[2] = x.z; d[3] = x.w;
  }
#endif
}

__device__ __forceinline__ void stage_wait_prev() {
#if USE_ASYNC_LDS
  __builtin_amdgcn_s_wait_asynccnt(8);  // keep only the 8 newest outstanding
#endif
}
__device__ __forceinline__ void stage_wait_all() {
#if USE_ASYNC_LDS
  __builtin_amdgcn_s_wait_asynccnt(0);
#endif
}

// ---------------------------------------------------------------------------
// Phase 1: Mpart[head,seg] = K[head, seg-range]^T @ V[head, seg-range]
// One block per (head, seg). Wave w owns tiles (d1 = w>>1, d2 = (w&1)*2..+1).
// ---------------------------------------------------------------------------
template <int NS>
__global__ __launch_bounds__(256) void sattn_ktv_kernel(
    const float* __restrict__ Kp, const float* __restrict__ Vp,
    float* __restrict__ Mpart) {
  __shared__ float kbuf[2][64 * KSTR];
  __shared__ float vbuf[2][64 * KSTR];

  const int tid  = threadIdx.x;
  const int wave = tid >> 5;
  const int lane = tid & 31;
  const int lh   = lane >> 4;
  const int ll   = lane & 15;
  const int d1b  = (wave >> 1) * 16;
  const int d2b  = (wave & 1) * 32;

  const int head = blockIdx.x / NS;
  const int seg  = blockIdx.x % NS;
  const int nch  = (SEQ / NS) / 64;  // 64-row chunks in this segment
  const size_t base = (size_t)head * SEQ * DIM + (size_t)seg * (SEQ / NS) * DIM;
  const float* kg = Kp + base;
  const float* vg = Vp + base;

  stage_issue(kg, kbuf[0], tid);
  stage_issue(vg, vbuf[0], tid);

  v8f c0 = {};
  v8f c1 = {};
  int cur = 0;
  for (int ch = 0; ch < nch; ++ch) {
    if (ch + 1 < nch) {
      stage_issue(kg + (size_t)(ch + 1) * 64 * DIM, kbuf[cur ^ 1], tid);
      stage_issue(vg + (size_t)(ch + 1) * 64 * DIM, vbuf[cur ^ 1], tid);
      stage_wait_prev();   // chunk ch complete; chunk ch+1 flies during compute
    } else {
      stage_wait_all();
    }
    __syncthreads();

    const float* kb = kbuf[cur];
    const float* vb = vbuf[cur];
#pragma unroll
    for (int k0 = 0; k0 < 64; k0 += 4) {
      const int r = k0 + 2 * lh;
      v2f a, b0, b1;
      a.x  = kb[(r    ) * KSTR + d1b + ll];      // A = K^T (m=d1, k=t)
      a.y  = kb[(r + 1) * KSTR + d1b + ll];
      b0.x = vb[(r    ) * KSTR + d2b + ll];      // B = V (k=t, n=d2)
      b0.y = vb[(r + 1) * KSTR + d2b + ll];
      b1.x = vb[(r    ) * KSTR + d2b + 16 + ll];
      b1.y = vb[(r + 1) * KSTR + d2b + 16 + ll];
      c0 = wmma4_f32(a, b0, c0);
      c1 = wmma4_f32(a, b1, c1);
    }
    __syncthreads();       // all waves done reading buf[cur] before reuse
    cur ^= 1;
  }

  float* mout = Mpart + (size_t)blockIdx.x * (DIM * DIM);  // unscaled partial
#pragma unroll
  for (int i = 0; i < 8; ++i) {
    const int row = d1b + i + 8 * lh;            // C/D layout: vgpr i = rows i, i+8
    mout[row * DIM + d2b + ll]      = c0[i];
    mout[row * DIM + d2b + 16 + ll] = c1[i];
  }
}

// ---------------------------------------------------------------------------
// Phase 2: out[head, s, :] = Q[head, s, :] @ (scale * sum_seg Mpart[head,seg])
// 2048 blocks: 64 rows of Q each; partial reduction fused into M staging.
// ---------------------------------------------------------------------------
template <int NS>
__global__ __launch_bounds__(256) void sattn_qm_kernel(
    const float* __restrict__ Qp, const float* __restrict__ Mpart,
    float* __restrict__ Op) {
  __shared__ float qbuf[64 * LSTR];
  __shared__ float mbuf[64 * LSTR];

  const int tid  = threadIdx.x;
  const int wave = tid >> 5;
  const int lane = tid & 31;
  const int lh   = lane >> 4;
  const int ll   = lane & 15;
  const int sb   = (wave >> 1) * 16;
  const int db   = (wave & 1) * 32;

  const int head = blockIdx.x >> 5;
  const int s0   = (blockIdx.x & 31) * 64;
  const size_t base = (size_t)head * SEQ * DIM + (size_t)s0 * DIM;

  {
    const float4* g4 = (const float4*)(Qp + base);
    const float4* m4 = (const float4*)(Mpart + (size_t)head * NS * (DIM * DIM));
    const float scale = 0.125f;  // 1/sqrt(64)
#pragma unroll
    for (int i = 0; i < 4; ++i) {
      const int e   = tid + i * 256;
      const int row = e >> 4;
      const int c4  = (e & 15) * 4;
      float4 x = g4[e];
      float* qd = &qbuf[row * LSTR + c4];
      qd[0] = x.x; qd[1] = x.y; qd[2] = x.z; qd[3] = x.w;
      float4 acc = m4[e];                       // seg 0
#pragma unroll
      for (int s = 1; s < NS; ++s) {            // fixed order -> deterministic
        float4 t = m4[(size_t)s * 1024 + e];
        acc.x += t.x; acc.y += t.y; acc.z += t.z; acc.w += t.w;
      }
      float* md = &mbuf[row * LSTR + c4];
      md[0] = acc.x * scale; md[1] = acc.y * scale;
      md[2] = acc.z * scale; md[3] = acc.w * scale;
    }
  }
  __syncthreads();

  v8f c0 = {};
  v8f c1 = {};
#pragma unroll
  for (int k0 = 0; k0 < 64; k0 += 4) {
    const int r = k0 + 2 * lh;
    v2f a, b0, b1;
    a.x  = qbuf[(sb + ll) * LSTR + r];          // A = Q (m=s, k); stride 65 ->
    a.y  = qbuf[(sb + ll) * LSTR + r + 1];      // lanes hit distinct banks
    b0.x = mbuf[(r    ) * LSTR + db + ll];      // B = M (k, n=d)
    b0.y = mbuf[(r + 1) * LSTR + db + ll];
    b1.x = mbuf[(r    ) * LSTR + db + 16 + ll];
    b1.y = mbuf[(r + 1) * LSTR + db + 16 + ll];
    c0 = wmma4_f32(a, b0, c0);
    c1 = wmma4_f32(a, b1, c1);
  }

  float* o = Op + base;
#pragma unroll
  for (int i = 0; i < 8; ++i) {
    const int row = sb + i + 8 * lh;
    o[row * DIM + db + ll]      = c0[i];
    o[row * DIM + db + 16 + ll] = c1[i];
  }
}

// ---------------------------------------------------------------------------
extern "C" void kernel_launch(void* const* d_in, const int* in_sizes, int n_in,
                              void* d_out, int out_size, void* d_ws,
                              size_t ws_size, hipStream_t stream) {
  const float* q = (const float*)d_in[0];
  const float* k = (const float*)d_in[1];
  const float* v = (const float*)d_in[2];
  float* out = (float*)d_out;
  float* ws  = (float*)d_ws;

  const size_t need8 = (size_t)NHEAD * 8 * DIM * DIM * sizeof(float);  // 8 MB
  if (ws_size >= need8) {
    sattn_ktv_kernel<8><<<NHEAD * 8, 256, 0, stream>>>(k, v, ws);
    sattn_qm_kernel<8><<<NHEAD * (SEQ / 64), 256, 0, stream>>>(q, ws, out);
  } else {
    sattn_ktv_kernel<1><<<NHEAD, 256, 0, stream>>>(k, v, ws);
    sattn_qm_kernel<1><<<NHEAD * (SEQ / 64), 256, 0, stream>>>(q, ws, out);
  }
}